// GumbelSinkhornStraightThrough_69715909149172
// MI455X (gfx1250) — compile-verified
//
#include <hip/hip_runtime.h>
#include <stdint.h>

typedef float v2f __attribute__((ext_vector_type(2)));
typedef float v8f __attribute__((ext_vector_type(8)));

#define SCALE   14.426950408889634f   /* log2(e) / TEMP, TEMP = 0.1 */
#define SFLOOR  1e-30f
#define NITER   20

// ---------------------------------------------------------------------------
// Kernel 0: sgS[v,L] = sigmoid(gamma[i(v,L), j(v,L)]) * SCALE, stored so the
// main kernel can load it with coalesced b128 loads straight into the
// WMMA A-layout register file.
// Layout: reg v = ((ti*4+tj)*4+c)*2+p ; lane L: i=16*ti+(L&15), j=16*tj+4c+p+2*(L>>4)
// Storage index: (v>>2)*128 + L*4 + (v&3)
// ---------------------------------------------------------------------------
__global__ __launch_bounds__(256) void gs_sigmoid_kernel(
    const float* __restrict__ gamma, float* __restrict__ sgS) {
  int tid = blockIdx.x * 256 + threadIdx.x;   // 0..4095
  int v = tid >> 5, L = tid & 31;
  int p  = v & 1;
  int c  = (v >> 1) & 3;
  int tj = (v >> 3) & 3;
  int ti = (v >> 5) & 3;
  int i = 16 * ti + (L & 15);
  int j = 16 * tj + 4 * c + p + 2 * (L >> 4);
  float x = gamma[i * 64 + j];
  float s = 1.0f / (1.0f + expf(-x));
  sgS[(v >> 2) * 128 + L * 4 + (v & 3)] = s * SCALE;
}

// ---------------------------------------------------------------------------
// Main kernel: one wave32 per sample. lx0 kept in registers (A-layout).
// Duals: Rr[ti] (row i = 16*ti + (L&15)), Cd[tj][c][p] (this lane's 32 cols).
// Row pass sums via v_wmma_f32_16x16x4_f32 with a ones B-matrix.
// Col pass sums via xor-butterfly within each 16-lane half.
// ---------------------------------------------------------------------------
__global__ __launch_bounds__(32) void gs_sinkhorn_kernel(
    const float* __restrict__ noise, const float* __restrict__ sgS,
    float* __restrict__ out) {
  __shared__ __align__(16) float lds[64 * 68];   // 68-float row pitch: no bank conflicts
  const int L = (int)(threadIdx.x & 31);
  const int b = (int)blockIdx.x;

  // ---- async DMA: noise[b] (row-major, 16KB) -> padded LDS -----------------
  {
    uint32_t laddr = (uint32_t)(uintptr_t)(&lds[0])
                   + (uint32_t)((L >> 4) * 272 + (L & 15) * 16);
    uint64_t gaddr = (uint64_t)(uintptr_t)(noise + (size_t)b * 4096)
                   + (uint64_t)(L * 16);
#pragma unroll
    for (int k = 0; k < 32; ++k) {   // 32 x (32 lanes x 16B) = 16KB
      asm volatile("global_load_async_to_lds_b128 %0, %1, off"
                   :: "v"(laddr), "v"(gaddr) : "memory");
      laddr += 544;   // 2 rows * 272B
      gaddr += 512;   // 2 rows * 256B
    }
    asm volatile("s_wait_asynccnt 0" ::: "memory");
  }

  // ---- build lx0 in A-layout registers ------------------------------------
  float lx0[4][4][4][2];
  {
    const float4* sg4 = (const float4*)sgS;
#pragma unroll
    for (int q = 0; q < 32; ++q) {
      float4 g = sg4[q * 32 + L];            // regs v = 4q..4q+3
      int c0 = (2 * q) & 3;                  // c of v=4q (pairs c0, c0+1)
      int tj = (q >> 1) & 3;
      int ti = (q >> 3) & 3;
      int i  = 16 * ti + (L & 15);
      int j0 = 16 * tj + 4 * c0 + 2 * (L >> 4);
      float2 n0 = *(const float2*)&lds[i * 68 + j0];
      float2 n1 = *(const float2*)&lds[i * 68 + j0 + 4];
      lx0[ti][tj][c0    ][0] = fmaf(n0.x, SCALE, g.x);
      lx0[ti][tj][c0    ][1] = fmaf(n0.y, SCALE, g.y);
      lx0[ti][tj][c0 + 1][0] = fmaf(n1.x, SCALE, g.z);
      lx0[ti][tj][c0 + 1][1] = fmaf(n1.y, SCALE, g.w);
    }
  }

  // ---- duals --------------------------------------------------------------
  float Cd[4][4][2];
  float Rr[4];
#pragma unroll
  for (int tj = 0; tj < 4; ++tj)
#pragma unroll
    for (int c = 0; c < 4; ++c) { Cd[tj][c][0] = 0.f; Cd[tj][c][1] = 0.f; }

  // Rr initialized to true row max (only the first row pass needs it).
#pragma unroll
  for (int ti = 0; ti < 4; ++ti) {
    float m = -3.4e38f;
#pragma unroll
    for (int tj = 0; tj < 4; ++tj)
#pragma unroll
      for (int c = 0; c < 4; ++c) {
        m = fmaxf(m, lx0[ti][tj][c][0]);
        m = fmaxf(m, lx0[ti][tj][c][1]);
      }
    m = fmaxf(m, __shfl_xor(m, 16, 32));   // other half holds the other 32 cols
    Rr[ti] = m;
  }

  const bool b0 = (L & 1) != 0, b1 = (L & 2) != 0, b2 = (L & 4) != 0;
  const bool hswap = (((L >> 3) & 1) != ((L >> 4) & 1));
  const v2f vones = {1.0f, 1.0f};

  // ---- row pass: R_i = Rr_i + log2( sum_j 2^(lx0 - Cd_j - Rr_i) ) ----------
  auto row_pass = [&]() {
#if __has_builtin(__builtin_amdgcn_wmma_f32_16x16x4_f32)
    v8f acc[4];
#pragma unroll
    for (int ti = 0; ti < 4; ++ti) acc[ti] = (v8f){0, 0, 0, 0, 0, 0, 0, 0};
#pragma unroll
    for (int ti = 0; ti < 4; ++ti)
#pragma unroll
      for (int tj = 0; tj < 4; ++tj)
#pragma unroll
        for (int c = 0; c < 4; ++c) {
          v2f a;
          a[0] = __builtin_amdgcn_exp2f(lx0[ti][tj][c][0] - Cd[tj][c][0] - Rr[ti]);
          a[1] = __builtin_amdgcn_exp2f(lx0[ti][tj][c][1] - Cd[tj][c][1] - Rr[ti]);
          // D = A(16x4) * ones(4x16) + C : row sums broadcast over N
          acc[ti] = __builtin_amdgcn_wmma_f32_16x16x4_f32(
              false, a, false, vones, (short)0, acc[ti], false, false);
        }
#pragma unroll
    for (int ti = 0; ti < 4; ++ti) {
      // pick component (L&7); fix half if bit3 != bit4 via xor16
      float x01 = b0 ? acc[ti][1] : acc[ti][0];
      float x23 = b0 ? acc[ti][3] : acc[ti][2];
      float x45 = b0 ? acc[ti][5] : acc[ti][4];
      float x67 = b0 ? acc[ti][7] : acc[ti][6];
      float y0 = b1 ? x23 : x01;
      float y1 = b1 ? x67 : x45;
      float z  = b2 ? y1 : y0;
      float o  = __shfl_xor(z, 16, 32);
      float s  = hswap ? o : z;
      Rr[ti] += __builtin_amdgcn_logf(fmaxf(s, SFLOOR));
    }
#else
#pragma unroll
    for (int ti = 0; ti < 4; ++ti) {
      float s = 0.f;
#pragma unroll
      for (int tj = 0; tj < 4; ++tj)
#pragma unroll
        for (int c = 0; c < 4; ++c) {
          s += __builtin_amdgcn_exp2f(lx0[ti][tj][c][0] - Cd[tj][c][0] - Rr[ti]);
          s += __builtin_amdgcn_exp2f(lx0[ti][tj][c][1] - Cd[tj][c][1] - Rr[ti]);
        }
      s += __shfl_xor(s, 16, 32);
      Rr[ti] += __builtin_amdgcn_logf(fmaxf(s, SFLOOR));
    }
#endif
  };

  // ---- col pass: C_j = Cd_j + log2( sum_i 2^(lx0 - Rr_i - Cd_j) ) ----------
  auto col_pass = [&]() {
    float part[4][4][2];
#pragma unroll
    for (int tj = 0; tj < 4; ++tj)
#pragma unroll
      for (int c = 0; c < 4; ++c) { part[tj][c][0] = 0.f; part[tj][c][1] = 0.f; }
#pragma unroll
    for (int ti = 0; ti < 4; ++ti)
#pragma unroll
      for (int tj = 0; tj < 4; ++tj)
#pragma unroll
        for (int c = 0; c < 4; ++c) {
          part[tj][c][0] += __builtin_amdgcn_exp2f(lx0[ti][tj][c][0] - Rr[ti] - Cd[tj][c][0]);
          part[tj][c][1] += __builtin_amdgcn_exp2f(lx0[ti][tj][c][1] - Rr[ti] - Cd[tj][c][1]);
        }
#pragma unroll
    for (int tj = 0; tj < 4; ++tj)
#pragma unroll
      for (int c = 0; c < 4; ++c)
#pragma unroll
        for (int p = 0; p < 2; ++p) {
          float s = part[tj][c][p];      // 16-lane butterfly within half
          s += __shfl_xor(s, 1, 32);
          s += __shfl_xor(s, 2, 32);
          s += __shfl_xor(s, 4, 32);
          s += __shfl_xor(s, 8, 32);
          Cd[tj][c][p] += __builtin_amdgcn_logf(fmaxf(s, SFLOOR));
        }
  };

  // ---- Sinkhorn loop: 20 row passes, 20 col passes, ends on a col pass ----
  row_pass();
#pragma unroll 1
  for (int t = 0; t < NITER - 1; ++t) { col_pass(); row_pass(); }
  col_pass();

  // ---- output: exp(lx) = 2^(lx0 - R - C) ----------------------------------
  float* ob = out + (size_t)b * 4096;
#pragma unroll
  for (int ti = 0; ti < 4; ++ti)
#pragma unroll
    for (int tj = 0; tj < 4; ++tj)
#pragma unroll
      for (int c = 0; c < 4; ++c) {
        float2 r;
        r.x = __builtin_amdgcn_exp2f(lx0[ti][tj][c][0] - Rr[ti] - Cd[tj][c][0]);
        r.y = __builtin_amdgcn_exp2f(lx0[ti][tj][c][1] - Rr[ti] - Cd[tj][c][1]);
        int i  = 16 * ti + (L & 15);
        int j0 = 16 * tj + 4 * c + 2 * (L >> 4);
        *(float2*)(ob + i * 64 + j0) = r;
      }
}

// ---------------------------------------------------------------------------
extern "C" void kernel_launch(void* const* d_in, const int* in_sizes, int n_in,
                              void* d_out, int out_size, void* d_ws, size_t ws_size,
                              hipStream_t stream) {
  const float* gamma = (const float*)d_in[0];   // [64,64] f32
  const float* noise = (const float*)d_in[1];   // [B,64,64] f32
  float* sgS = (float*)d_ws;                    // 4096 floats scratch
  float* outp = (float*)d_out;

  int nsamples = in_sizes[1] / 4096;

  gs_sigmoid_kernel<<<16, 256, 0, stream>>>(gamma, sgS);
  gs_sinkhorn_kernel<<<nsamples, 32, 0, stream>>>(noise, sgS, outp);
}